// Source_Net_34248069218570
// MI455X (gfx1250) — compile-verified
//
#include <hip/hip_runtime.h>
#include <hip/hip_bf16.h>

#define KDIM 128
#define LOWD 64
#define PPD 64
#define NBATCH 4096
#define NSTU 50000
#define KPAD 129                 // 128 + 1 pad -> conflict-free column walks
#define ROWS_PER_BLOCK 16

typedef float v2f __attribute__((ext_vector_type(2)));
typedef float v8f __attribute__((ext_vector_type(8)));

// sigmoid via hardware transcendentals only: v_exp_f32 + v_rcp_f32
// (avoids the ~8-op IEEE v_div_scale/div_fixup sequence; ~1 ulp rcp)
__device__ __forceinline__ float sigf(float x) {
    return __builtin_amdgcn_rcpf(1.0f + __expf(-x));
}

// ---------------------------------------------------------------------------
// Stage 1: gather per-batch input rows.
//   Astu[b, 0:64]   = student_emb.reshape(NR*STU, LOW)[stu_id[b]]
//   Aexr[b, 0:64]   = sig(k_difficulty[exer_id[b]])
//   CATstu[b, 0:64] = prompt_stu[stu_id[b] % STU]   (cols 64:192 filled by GEMM1)
//   CATexr[b, 0:64] = sig(s_exer_vectors[region(exer_id[b])])
// ---------------------------------------------------------------------------
__global__ void gather_kernel(const int* __restrict__ stu_id,
                              const int* __restrict__ exer_id,
                              const float* __restrict__ student_emb,
                              const float* __restrict__ prompt_stu,
                              const float* __restrict__ k_difficulty,
                              const float* __restrict__ s_exer_vectors,
                              float* __restrict__ Astu,
                              float* __restrict__ Aexr,
                              float* __restrict__ CATstu,
                              float* __restrict__ CATexr)
{
    int idx = blockIdx.x * blockDim.x + threadIdx.x;
    if (idx >= NBATCH * LOWD) return;
    int b = idx >> 6;
    int l = idx & 63;
    int sid = stu_id[b];
    int eid = exer_id[b];
    Astu[idx] = student_emb[(long)sid * LOWD + l];
    Aexr[idx] = sigf(k_difficulty[(long)eid * LOWD + l]);
    CATstu[b * 192 + l] = prompt_stu[(long)(sid % NSTU) * PPD + l];
    int r = (eid <= 9999) ? 0 : 1;     // counts = [10000, 10000]
    CATexr[b * 192 + l] = sigf(s_exer_vectors[r * PPD + l]);
}

// ---------------------------------------------------------------------------
// Tiny constant GEMMs: KP1[k,j] = sum_l kle[k,l]*W1[j, K+l]; same for KP2/W2.
// ---------------------------------------------------------------------------
__global__ void kp_kernel(const float* __restrict__ kle,
                          const float* __restrict__ W1,
                          const float* __restrict__ W2,
                          float* __restrict__ KP1,
                          float* __restrict__ KP2)
{
    int idx = blockIdx.x * blockDim.x + threadIdx.x;
    if (idx >= 2 * KDIM * KDIM) return;
    int sel = idx >> 14;                   // 0 -> KP1, 1 -> KP2
    int rem = idx & (KDIM * KDIM - 1);
    int k = rem >> 7;
    int j = rem & 127;
    const float* W = sel ? W2 : W1;
    const float* kr = kle + k * LOWD;
    const float* wr = W + j * (KDIM + LOWD) + KDIM;   // W?b row j
    float acc = 0.f;
#pragma unroll 8
    for (int l = 0; l < LOWD; ++l) acc += kr[l] * wr[l];
    (sel ? KP2 : KP1)[rem] = acc;
}

// ---------------------------------------------------------------------------
// f32 WMMA GEMM, compile-time specialized:
//   C[m,n] = act( sum_k A[m,k] * W[n,k] (+ bias[n]) )
// Block = 128 threads = 4 waves; wave w owns output tile (m0 + 16*w, n0),
// computed with V_WMMA_F32_16X16X4_F32 (exact f32 math on the matrix pipe).
//
// ISA VGPR layouts (wave32):
//   A 16x4 : vgpr v, lane l: A[l&15][2*(l>>4) + v]
//   B 4x16 : vgpr v, lane l: B[2*(l>>4) + v][l&15]   (B[k][n] = W[n0+n][k0+k])
//   C 16x16: vgpr r, lane l: C[r + 8*(l>>4)][l&15]
// ---------------------------------------------------------------------------
template <int Kd, bool SIG, bool BIAS>
__global__ void wmma_gemm_f32(const float* __restrict__ A, int lda,
                              const float* __restrict__ W, int ldw,
                              const float* __restrict__ bias,
                              float* __restrict__ C, int ldc)
{
    const int wid  = threadIdx.x >> 5;               // wave id 0..3 (uniform)
    const int lane = threadIdx.x & 31;
    const int m0   = (blockIdx.x * 4 + wid) * 16;
    const int n0   = blockIdx.y * 16;
    const int half = lane >> 4;
    const int mr   = lane & 15;

    const float* arow = A + (long)(m0 + mr) * lda;
    const float* wrow = W + (long)(n0 + mr) * ldw;

    v8f acc = {};
#pragma unroll 8
    for (int k0 = 0; k0 < Kd; k0 += 4) {
        v2f a, b;
        a.x = arow[k0 + 2 * half + 0];
        a.y = arow[k0 + 2 * half + 1];
        b.x = wrow[k0 + 2 * half + 0];
        b.y = wrow[k0 + 2 * half + 1];
        acc = __builtin_amdgcn_wmma_f32_16x16x4_f32(
            /*neg_a=*/false, a, /*neg_b=*/false, b,
            /*c_mod=*/(short)0, acc, /*reuse_a=*/false, /*reuse_b=*/false);
    }

    const int col = n0 + mr;
    float bv = 0.0f;
    if (BIAS) bv = bias[col];
#pragma unroll
    for (int r = 0; r < 8; ++r) {
        float v = acc[r] + bv;
        if (SIG) v = sigf(v);
        C[(long)(m0 + r + 8 * half) * ldc + col] = v;
    }
}

// ---------------------------------------------------------------------------
// Final stage: out[b] = sum_k kn[b,k]*sig( sum_j (sig(SP[b,j]+KP1[k,j])
//                        - sig(EP[b,j]+KP2[k,j]))*W3[j] + b3 ) / sum_k kn[b,k]
//
// Block = 128 threads (thread t owns k = t), processes ROWS_PER_BLOCK batch
// rows. KP1/KP2 (128KB) staged once per block into LDS with +1 row padding:
// in the j-loop, lane k reads address k*129+j -> bank (k+j)%64, conflict-free.
// Uses CDNA5's 320KB/WGP LDS (134KB/block).
// ---------------------------------------------------------------------------
__global__ void final_kernel(const float* __restrict__ SP,
                             const float* __restrict__ EP,
                             const float* __restrict__ KP1,
                             const float* __restrict__ KP2,
                             const float* __restrict__ W3,
                             const float* __restrict__ b3,
                             const float* __restrict__ kn_emb,
                             float* __restrict__ out)
{
    __shared__ float sKP1[KDIM * KPAD];
    __shared__ float sKP2[KDIM * KPAD];
    __shared__ float sW3[KDIM];
    __shared__ float sSP[KDIM], sEP[KDIM];
    __shared__ float red[KDIM], redkn[KDIM];

    const int k = threadIdx.x;           // 0..127

    // Coalesced cooperative staging of KP1/KP2 into padded LDS.
#pragma unroll 4
    for (int idx = k; idx < KDIM * KDIM; idx += KDIM) {
        int r = idx >> 7, c = idx & 127;
        sKP1[r * KPAD + c] = KP1[idx];
        sKP2[r * KPAD + c] = KP2[idx];
    }
    sW3[k] = W3[k];
    __syncthreads();

    const float bb3 = b3[0];
    const float* kp1 = sKP1 + k * KPAD;
    const float* kp2 = sKP2 + k * KPAD;

    for (int rb = 0; rb < ROWS_PER_BLOCK; ++rb) {
        const int b = blockIdx.x * ROWS_PER_BLOCK + rb;
        sSP[k] = SP[b * KDIM + k];
        sEP[k] = EP[b * KDIM + k];
        __syncthreads();

        float acc = 0.f;
#pragma unroll 4
        for (int j = 0; j < KDIM; ++j) {
            float p = sigf(sSP[j] + kp1[j]);
            float d = sigf(sEP[j] + kp2[j]);
            acc = __builtin_fmaf(p - d, sW3[j], acc);
        }
        float o  = sigf(acc + bb3);
        float kn = kn_emb[b * KDIM + k];
        red[k]   = o * kn;
        redkn[k] = kn;
        __syncthreads();
#pragma unroll
        for (int s = 64; s > 0; s >>= 1) {
            if (k < s) { red[k] += red[k + s]; redkn[k] += redkn[k + s]; }
            __syncthreads();
        }
        if (k == 0) out[b] = red[0] / redkn[0];
        __syncthreads();                 // protect sSP/red before next row
    }
}

// ---------------------------------------------------------------------------
// Orchestration
// ---------------------------------------------------------------------------
extern "C" void kernel_launch(void* const* d_in, const int* in_sizes, int n_in,
                              void* d_out, int out_size, void* d_ws, size_t ws_size,
                              hipStream_t stream) {
    (void)in_sizes; (void)n_in; (void)out_size; (void)ws_size;

    const int*   stu_id      = (const int*)  d_in[0];
    const int*   exer_id     = (const int*)  d_in[1];
    const float* kn_emb      = (const float*)d_in[2];
    const float* student_emb = (const float*)d_in[3];
    const float* prompt_stu  = (const float*)d_in[4];
    const float* kle         = (const float*)d_in[5];
    const float* k_diff      = (const float*)d_in[6];
    const float* s_exer      = (const float*)d_in[7];
    const float* W1          = (const float*)d_in[8];
    const float* W2          = (const float*)d_in[9];
    const float* W3          = (const float*)d_in[10];
    const float* b3          = (const float*)d_in[11];
    const float* fc1_W       = (const float*)d_in[12];
    const float* fc1_b       = (const float*)d_in[13];
    const float* fc2_W       = (const float*)d_in[14];
    const float* fc2_b       = (const float*)d_in[15];
    float* outp = (float*)d_out;

    // Workspace layout
    float* ws = (float*)d_ws;
    float* Astu   = ws;                       // B*64
    float* Aexr   = Astu   + NBATCH * LOWD;   // B*64
    float* CATstu = Aexr   + NBATCH * LOWD;   // B*192
    float* CATexr = CATstu + NBATCH * 192;    // B*192
    float* STUE   = CATexr + NBATCH * 192;    // B*128
    float* EXRE   = STUE   + NBATCH * KDIM;   // B*128
    float* SP     = EXRE   + NBATCH * KDIM;   // B*128
    float* EP     = SP     + NBATCH * KDIM;   // B*128
    float* KP1    = EP     + NBATCH * KDIM;   // 128*128
    float* KP2    = KP1    + KDIM * KDIM;     // 128*128

    // Stage 1: gathers + small sigmoids
    gather_kernel<<<(NBATCH * LOWD + 255) / 256, 256, 0, stream>>>(
        stu_id, exer_id, student_emb, prompt_stu, k_diff, s_exer,
        Astu, Aexr, CATstu, CATexr);

    // Constant 128x128 GEMMs KP1/KP2
    kp_kernel<<<(2 * KDIM * KDIM + 255) / 256, 256, 0, stream>>>(
        kle, W1, W2, KP1, KP2);

    dim3 gemm_grid(NBATCH / 64, KDIM / 16);   // 4 M-tiles per block (4 waves)
    dim3 gemm_blk(128);

    // GEMM 1: old_stu / old_exer -> sigmoid -> cols 64:192 of concat buffers
    wmma_gemm_f32<LOWD, true, false><<<gemm_grid, gemm_blk, 0, stream>>>(
        Astu, LOWD, kle, LOWD, nullptr, CATstu + LOWD, 192);
    wmma_gemm_f32<LOWD, true, false><<<gemm_grid, gemm_blk, 0, stream>>>(
        Aexr, LOWD, kle, LOWD, nullptr, CATexr + LOWD, 192);

    // GEMM 2: stu_e / exer_e = sig(concat @ fcW^T + fcb)
    wmma_gemm_f32<192, true, true><<<gemm_grid, gemm_blk, 0, stream>>>(
        CATstu, 192, fc1_W, 192, fc1_b, STUE, KDIM);
    wmma_gemm_f32<192, true, true><<<gemm_grid, gemm_blk, 0, stream>>>(
        CATexr, 192, fc2_W, 192, fc2_b, EXRE, KDIM);

    // GEMM 3: SP = stu_e @ W1a^T, EP = exer_e @ W2a^T (W?a = first 128 cols
    // of the 192-wide W1/W2, so ldw = 192; no sigmoid)
    wmma_gemm_f32<KDIM, false, false><<<gemm_grid, gemm_blk, 0, stream>>>(
        STUE, KDIM, W1, KDIM + LOWD, nullptr, SP, KDIM);
    wmma_gemm_f32<KDIM, false, false><<<gemm_grid, gemm_blk, 0, stream>>>(
        EXRE, KDIM, W2, KDIM + LOWD, nullptr, EP, KDIM);

    // Final reduction over (k, j): 256 blocks x 16 rows each
    final_kernel<<<NBATCH / ROWS_PER_BLOCK, KDIM, 0, stream>>>(
        SP, EP, KP1, KP2, W3, b3, kn_emb, outp);
}